// HyperGCN_188978561647
// MI455X (gfx1250) — compile-verified
//
#include <hip/hip_runtime.h>
#include <math.h>

typedef __attribute__((ext_vector_type(2))) float v2f;
typedef __attribute__((ext_vector_type(8))) float v8f;

#define D_DIM 128
#define K_DIM 8
#define H1_DIM 8
#define C_DIM 16

// Guaranteed native L2 atomic add (non-returning, STOREcnt-tracked).
__device__ __forceinline__ void atom_add_f32(float* p, float v) {
    asm volatile("global_atomic_add_f32 %0, %1, off scope:SCOPE_DEV"
                 :: "v"(p), "v"(v) : "memory");
}

__global__ void k_init_deg(float* deg, int V) {
    int v = blockIdx.x * blockDim.x + threadIdx.x;
    if (v < V) deg[v] = 1.0f;
}

// Combined B matrix [128 x 16]: cols 0..7 = W1, col 8 = rv, cols 9..15 = 0
__global__ void k_build_B(const float* __restrict__ W1, const float* __restrict__ rv,
                          float* __restrict__ Bc) {
    int idx = blockIdx.x * blockDim.x + threadIdx.x;
    if (idx < D_DIM * 16) {
        int k = idx >> 4, n = idx & 15;
        float val = 0.0f;
        if (n < H1_DIM)       val = W1[k * H1_DIM + n];
        else if (n == H1_DIM) val = rv[k];
        Bc[idx] = val;
    }
}

// Fused proj = X@rv and XW1 = X@W1 via V_WMMA_F32_16X16X4_F32.
// One wave computes one 16-row tile; 32 WMMA steps cover K=128.
__global__ void k_wmma_xw1(const float* __restrict__ X,
                           const float* __restrict__ Bc,
                           float* __restrict__ XW1,
                           float* __restrict__ proj,
                           int ntiles) {
    int lane = threadIdx.x & 31;
    int wave = threadIdx.x >> 5;
    int tile = blockIdx.x * 4 + wave;
    if (tile >= ntiles) return;          // whole-wave uniform: EXEC stays all-1s
    int half = lane >> 4;                // 0: K pair (0,1); 1: K pair (2,3)
    int n    = lane & 15;                // B/D column, and A row (M)

    // B fragments for all 32 K-steps (compiler pipelines these loads)
    v2f b[32];
#pragma unroll
    for (int kk = 0; kk < 32; ++kk) {
        int k = kk * 4 + 2 * half;
        b[kk].x = Bc[k * 16 + n];
        b[kk].y = Bc[(k + 1) * 16 + n];
    }

    const float* xrow = X + (size_t)(tile * 16 + n) * D_DIM;
    v8f acc = {};
#pragma unroll
    for (int kk = 0; kk < 32; ++kk) {
        v2f a = *(const v2f*)(xrow + kk * 4 + 2 * half);   // A: 16x4 f32 fragment
        acc = __builtin_amdgcn_wmma_f32_16x16x4_f32(
                  false, a, false, b[kk], (short)0, acc, false, false);
    }

    // D layout: VGPR r, lane l -> (M = r + 8*half, N = n)
#pragma unroll
    for (int r = 0; r < 8; ++r) {
        int row = tile * 16 + r + 8 * half;
        if (n < H1_DIM)       XW1[(size_t)row * H1_DIM + n] = acc[r];
        else if (n == H1_DIM) proj[row] = acc[r];
    }
}

__global__ void k_edge_select(const int* __restrict__ E, const float* __restrict__ proj,
                              int* __restrict__ Se, int* __restrict__ Ie,
                              float* deg, int NE) {
    int e = blockIdx.x * blockDim.x + threadIdx.x;
    if (e >= NE) return;
    const int4* Ee4 = (const int4*)(E + (size_t)e * K_DIM);
    int4 e0 = Ee4[0], e1 = Ee4[1];
    int idx[K_DIM] = {e0.x, e0.y, e0.z, e0.w, e1.x, e1.y, e1.z, e1.w};
    int sa = idx[0], si = idx[0];
    float pmax = proj[sa], pmin = pmax;
#pragma unroll
    for (int j = 1; j < K_DIM; ++j) {
        int v = idx[j];
        float p = proj[v];
        if (p > pmax) { pmax = p; sa = v; }   // strict: first-occurrence argmax
        if (p < pmin) { pmin = p; si = v; }   // strict: first-occurrence argmin
    }
    Se[e] = sa; Ie[e] = si;
    atom_add_f32(deg + sa, 1.0f / K_DIM);
    atom_add_f32(deg + si, 1.0f / K_DIM);
}

__global__ void k_rsqrt_inplace(float* deg, int V) {
    int v = blockIdx.x * blockDim.x + threadIdx.x;
    if (v < V) deg[v] = rsqrtf(deg[v]);
}

template<int F>
__global__ void k_diag(const float* __restrict__ M, const float* __restrict__ dinv,
                       float* __restrict__ Acc, int V) {
    int v = blockIdx.x * blockDim.x + threadIdx.x;
    if (v >= V) return;
    float s = dinv[v] * dinv[v];
    const float4* Mv = (const float4*)(M + (size_t)v * F);
    float4*       Av = (float4*)(Acc + (size_t)v * F);
#pragma unroll
    for (int q = 0; q < F / 4; ++q) {
        float4 m = Mv[q];
        Av[q] = make_float4(s * m.x, s * m.y, s * m.z, s * m.w);
    }
}

template<int F>
__global__ void k_scatter(const int* __restrict__ Se, const int* __restrict__ Ie,
                          const float* __restrict__ dinv, const float* __restrict__ M,
                          float* __restrict__ Acc, int NE) {
    int e = blockIdx.x * blockDim.x + threadIdx.x;
    if (e >= NE) return;
    int s = Se[e], i = Ie[e];
    float w = (1.0f / K_DIM) * dinv[s] * dinv[i];
    const float4* Ms4 = (const float4*)(M + (size_t)s * F);
    const float4* Mi4 = (const float4*)(M + (size_t)i * F);
    float* As = Acc + (size_t)s * F;
    float* Ai = Acc + (size_t)i * F;
#pragma unroll
    for (int q = 0; q < F / 4; ++q) {
        float4 ms = Ms4[q];
        float4 mi = Mi4[q];
        atom_add_f32(As + 4 * q + 0, w * mi.x);
        atom_add_f32(As + 4 * q + 1, w * mi.y);
        atom_add_f32(As + 4 * q + 2, w * mi.z);
        atom_add_f32(As + 4 * q + 3, w * mi.w);
        atom_add_f32(Ai + 4 * q + 0, w * ms.x);
        atom_add_f32(Ai + 4 * q + 1, w * ms.y);
        atom_add_f32(Ai + 4 * q + 2, w * ms.z);
        atom_add_f32(Ai + 4 * q + 3, w * ms.w);
    }
}

// H1 = relu(Acc1 + b1); M2 = H1 @ W2 (8x8); Acc2 = dinv^2 * M2
__global__ void k_layer2(const float* __restrict__ Acc1, const float* __restrict__ b1,
                         const float* __restrict__ W2, const float* __restrict__ dinv,
                         float* __restrict__ M2, float* __restrict__ Acc2, int V) {
    int v = blockIdx.x * blockDim.x + threadIdx.x;
    if (v >= V) return;
    float h[H1_DIM];
#pragma unroll
    for (int j = 0; j < H1_DIM; ++j)
        h[j] = fmaxf(Acc1[(size_t)v * H1_DIM + j] + b1[j], 0.0f);
    float dv2 = dinv[v] * dinv[v];
#pragma unroll
    for (int n = 0; n < H1_DIM; ++n) {
        float m = 0.0f;
#pragma unroll
        for (int j = 0; j < H1_DIM; ++j) m += h[j] * W2[j * H1_DIM + n];
        M2[(size_t)v * H1_DIM + n]  = m;
        Acc2[(size_t)v * H1_DIM + n] = dv2 * m;
    }
}

// H2 = relu(Acc2 + b2); M3 = H2 @ W3 (8x16); Acc3 = dinv^2 * M3
__global__ void k_layer3(const float* __restrict__ Acc2, const float* __restrict__ b2,
                         const float* __restrict__ W3, const float* __restrict__ dinv,
                         float* __restrict__ M3, float* __restrict__ Acc3, int V) {
    int v = blockIdx.x * blockDim.x + threadIdx.x;
    if (v >= V) return;
    float h[H1_DIM];
#pragma unroll
    for (int j = 0; j < H1_DIM; ++j)
        h[j] = fmaxf(Acc2[(size_t)v * H1_DIM + j] + b2[j], 0.0f);
    float dv2 = dinv[v] * dinv[v];
#pragma unroll
    for (int n = 0; n < C_DIM; ++n) {
        float m = 0.0f;
#pragma unroll
        for (int j = 0; j < H1_DIM; ++j) m += h[j] * W3[j * C_DIM + n];
        M3[(size_t)v * C_DIM + n]  = m;
        Acc3[(size_t)v * C_DIM + n] = dv2 * m;
    }
}

// H3 = relu(Acc3 + b3); out = sigmoid(H3 . fc_w + fc_b)
__global__ void k_final(const float* __restrict__ Acc3, const float* __restrict__ b3,
                        const float* __restrict__ fcw, const float* __restrict__ fcb,
                        float* __restrict__ out, int V) {
    int v = blockIdx.x * blockDim.x + threadIdx.x;
    if (v >= V) return;
    float logit = fcb[0];
#pragma unroll
    for (int n = 0; n < C_DIM; ++n) {
        float h = fmaxf(Acc3[(size_t)v * C_DIM + n] + b3[n], 0.0f);
        logit += h * fcw[n];
    }
    out[v] = 1.0f / (1.0f + expf(-logit));
}

static inline size_t alignup(size_t x) { return (x + 63) & ~(size_t)63; }

extern "C" void kernel_launch(void* const* d_in, const int* in_sizes, int n_in,
                              void* d_out, int out_size, void* d_ws, size_t ws_size,
                              hipStream_t stream) {
    const float* X    = (const float*)d_in[0];
    const int*   E    = (const int*)d_in[1];
    const float* rv   = (const float*)d_in[2];
    const float* W1   = (const float*)d_in[3];
    const float* b1   = (const float*)d_in[4];
    const float* W2   = (const float*)d_in[5];
    const float* b2   = (const float*)d_in[6];
    const float* W3   = (const float*)d_in[7];
    const float* b3   = (const float*)d_in[8];
    const float* fcw  = (const float*)d_in[9];
    const float* fcb  = (const float*)d_in[10];
    float* out = (float*)d_out;

    const int V  = in_sizes[0] / D_DIM;
    const int NE = in_sizes[1] / K_DIM;

    // Workspace layout (elements, 64-elem aligned regions)
    size_t off = 0;
    float* deg  = (float*)d_ws + off; off += alignup((size_t)V);        // deg -> dinv in place
    float* proj = (float*)d_ws + off; off += alignup((size_t)V);
    int*   Se   = (int*)((float*)d_ws + off); off += alignup((size_t)NE);
    int*   Ie   = (int*)((float*)d_ws + off); off += alignup((size_t)NE);
    float* XW1  = (float*)d_ws + off; off += alignup((size_t)V * H1_DIM); // reused as Acc2
    float* Acc1 = (float*)d_ws + off; off += alignup((size_t)V * H1_DIM);
    float* M2   = (float*)d_ws + off; off += alignup((size_t)V * H1_DIM);
    float* M3   = (float*)d_ws + off; off += alignup((size_t)V * C_DIM);
    float* Acc3 = (float*)d_ws + off; off += alignup((size_t)V * C_DIM);
    float* Bc   = (float*)d_ws + off; off += alignup((size_t)D_DIM * 16);
    float* Acc2 = XW1;   // XW1 dead after scatter-1

    const int T = 256;
    const int gV  = (V + T - 1) / T;
    const int gNE = (NE + T - 1) / T;

    k_init_deg<<<gV, T, 0, stream>>>(deg, V);
    k_build_B<<<(D_DIM * 16 + T - 1) / T, T, 0, stream>>>(W1, rv, Bc);

    int ntiles = (V + 15) / 16;
    int nblk   = (ntiles + 3) / 4;                  // 4 waves (128 threads) per block
    k_wmma_xw1<<<nblk, 128, 0, stream>>>(X, Bc, XW1, proj, ntiles);

    k_edge_select<<<gNE, T, 0, stream>>>(E, proj, Se, Ie, deg, NE);
    k_rsqrt_inplace<<<gV, T, 0, stream>>>(deg, V);  // deg now holds dinv

    // Layer 1: spmm(XW1) -> Acc1
    k_diag<H1_DIM><<<gV, T, 0, stream>>>(XW1, deg, Acc1, V);
    k_scatter<H1_DIM><<<gNE, T, 0, stream>>>(Se, Ie, deg, XW1, Acc1, NE);

    // Layer 2
    k_layer2<<<gV, T, 0, stream>>>(Acc1, b1, W2, deg, M2, Acc2, V);
    k_scatter<H1_DIM><<<gNE, T, 0, stream>>>(Se, Ie, deg, M2, Acc2, NE);

    // Layer 3
    k_layer3<<<gV, T, 0, stream>>>(Acc2, b2, W3, deg, M3, Acc3, V);
    k_scatter<C_DIM><<<gNE, T, 0, stream>>>(Se, Ie, deg, M3, Acc3, NE);

    // Head
    k_final<<<gV, T, 0, stream>>>(Acc3, b3, fcw, fcb, out, V);
}